// EmbracementLayer_79628693667963
// MI455X (gfx1250) — compile-verified
//
#include <hip/hip_runtime.h>
#include <hip/hip_bf16.h>
#include <cmath>

typedef __attribute__((ext_vector_type(2))) float v2f;
typedef __attribute__((ext_vector_type(8))) float v8f;

#define B_  64
#define S_  512
#define D_  1024
#define K_  2048   // 2*D

// ---------------------------------------------------------------------------
// Kernel 1: build Combined (64 x 2048) in workspace.
//   Combined[b, d]        = tokens[b, idx[d]+1, d]   (d < 1024)
//   Combined[b, 1024 + d] = cls_output[b, d]
// (softmax over a single key is identically 1, so mix == embraced)
// ---------------------------------------------------------------------------
__global__ void build_combined_kernel(const float* __restrict__ tokens,
                                      const float* __restrict__ cls,
                                      const int*   __restrict__ idx,
                                      float*       __restrict__ comb) {
    const int tid = blockIdx.x * blockDim.x + threadIdx.x;  // 0 .. 64*2048-1
    const int b = tid >> 11;        // / 2048
    const int c = tid & (K_ - 1);   // % 2048
    float v;
    if (c < D_) {
        const int s = idx[c] + 1;   // the [:,1:,:] slice shift
        v = tokens[((size_t)b * S_ + s) * D_ + c];
    } else {
        v = cls[(size_t)b * D_ + (c - D_)];
    }
    comb[(size_t)tid] = v;
}

// ---------------------------------------------------------------------------
// Kernel 2: out(64x1024) = tanh( Combined(64x2048) * W_out^T(2048x1024) + b )
// One wave32 per 16x16 output tile, f32 WMMA (V_WMMA_F32_16X16X4_F32).
//
// ISA 7.12.2 layouts (wave32):
//   A 16x4 f32 : lane l -> M = l%16,        K = kb + 2*(l/16) + {0,1}  (v2f)
//   B 4x16 f32 : lane l -> N = l%16,        K = kb + 2*(l/16) + {0,1}  (v2f)
//   C/D 16x16  : vgpr v, lane l -> M = v + 8*(l/16), N = l%16          (v8f)
// ---------------------------------------------------------------------------
__global__ void gemm_bias_tanh_kernel(const float* __restrict__ comb,  // 64x2048
                                      const float* __restrict__ Wout,  // 1024x2048 row-major
                                      const float* __restrict__ bout,  // 1024
                                      float*       __restrict__ out) { // 64x1024
    const int lane  = threadIdx.x;      // 0..31 (wave32, EXEC all ones)
    const int mt    = blockIdx.x;       // 0..3   -> batch tile
    const int nt    = blockIdx.y;       // 0..63  -> output-dim tile
    const int b0    = mt << 4;
    const int n0    = nt << 4;
    const int lmod  = lane & 15;
    const int lhalf = lane >> 4;        // 0 or 1

    // Per-lane base pointers (8-byte aligned -> global_load_b64)
    const float* aPtr = comb + (size_t)(b0 + lmod) * K_ + 2 * lhalf;
    const float* wPtr = Wout + (size_t)(n0 + lmod) * K_ + 2 * lhalf;  // W^T[k,n]=Wout[n,k]

    v8f acc = {};
#pragma unroll 4
    for (int kb = 0; kb < K_; kb += 4) {
        const v2f a = *(const v2f*)(aPtr + kb);
        const v2f w = *(const v2f*)(wPtr + kb);
        // 8 args: (neg_a, A, neg_b, B, c_mod, C, reuse_a, reuse_b)
        acc = __builtin_amdgcn_wmma_f32_16x16x4_f32(
                  false, a, false, w, (short)0, acc, false, false);
    }

    const int   n    = n0 + lmod;
    const float bias = bout[n];
#pragma unroll
    for (int v = 0; v < 8; ++v) {
        const int m = b0 + v + 8 * lhalf;
        out[(size_t)m * D_ + n] = tanhf(acc[v] + bias);
    }
}

// ---------------------------------------------------------------------------
// Host launcher
// ---------------------------------------------------------------------------
extern "C" void kernel_launch(void* const* d_in, const int* in_sizes, int n_in,
                              void* d_out, int out_size, void* d_ws, size_t ws_size,
                              hipStream_t stream) {
    const float* tokens = (const float*)d_in[0];   // (64, 512, 1024) f32
    const float* cls    = (const float*)d_in[1];   // (64, 1024) f32
    const float* Wout   = (const float*)d_in[2];   // (1024, 2048) f32
    const float* bout   = (const float*)d_in[3];   // (1024,) f32
    const int*   idx    = (const int*)  d_in[4];   // (1024,) i32
    float*       out    = (float*)d_out;           // (64, 1024) f32
    float*       comb   = (float*)d_ws;            // 64*2048 f32 = 512 KB scratch

    // Stage the gathered/concatenated activation matrix once.
    const int total = B_ * K_;                      // 131072
    build_combined_kernel<<<dim3(total / 256), 256, 0, stream>>>(tokens, cls, idx, comb);

    // WMMA GEMM + bias + tanh: grid = (M tiles, N tiles), 1 wave per tile.
    gemm_bias_tanh_kernel<<<dim3(B_ / 16, D_ / 16), 32, 0, stream>>>(comb, Wout, bout, out);
}